// CRF_47141561041240
// MI455X (gfx1250) — compile-verified
//
#include <hip/hip_runtime.h>
#include <hip/hip_bf16.h>
#include <stdint.h>

// CRF sequence score. Shapes fixed by the reference.
#define SEQ    512
#define BATCH  512
#define NTAGS  128
#define THREADS 256
#define SPT (SEQ / THREADS)   // 2 sequence steps per thread

__global__ __launch_bounds__(THREADS)
void crf_score_kernel(const float* __restrict__ emissions,  // (S,B,T) f32
                      const int*   __restrict__ tags,       // (S,B) i32
                      const int*   __restrict__ mask,       // (S,B) i32
                      const float* __restrict__ start_tr,   // (T)
                      const float* __restrict__ end_tr,     // (T)
                      const float* __restrict__ trans,      // (T,T)
                      float* __restrict__ out)              // (B)
{
    __shared__ int   sTags[SEQ];      // staged tag column for this batch element
    __shared__ float sSum[THREADS];
    __shared__ int   sCnt[THREADS];

    const int b = blockIdx.x;   // one block per batch column
    const int t = threadIdx.x;

    // ---- Stage this column's tags into LDS with gfx1250 async-to-LDS DMA ----
    // Each lane supplies its own global address and LDS destination offset;
    // completion tracked on ASYNCcnt (CDNA5 path, not DScnt/LOADcnt).
    #pragma unroll
    for (int k = 0; k < SPT; ++k) {
        const int s = t + k * THREADS;
        uint64_t gaddr  = (uint64_t)(tags + (size_t)s * BATCH + b);
        uint32_t ldsOff = (uint32_t)(size_t)(&sTags[s]);   // low 32 bits of generic LDS ptr = LDS byte addr
        asm volatile("global_load_async_to_lds_b32 %0, %1, off"
                     :: "v"(ldsOff), "v"(gaddr) : "memory");
    }
    asm volatile("s_wait_asynccnt 0" ::: "memory");  // my async writes landed in LDS
    __syncthreads();                                 // everyone else's too

    // ---- Per-thread partial sum over its sequence steps ----
    float acc = 0.0f;
    int   cnt = 0;
    #pragma unroll
    for (int k = 0; k < SPT; ++k) {
        const int s = t + k * THREADS;
        const int m = mask[(size_t)s * BATCH + b];
        cnt += m;
        if (s >= 1) {
            const int cur = sTags[s];
            const int prv = sTags[s - 1];
            const float term = trans[prv * NTAGS + cur]
                             + emissions[((size_t)s * BATCH + b) * NTAGS + cur];
            acc += (float)m * term;   // matches reference's mask-multiply semantics
        }
    }
    sSum[t] = acc;
    sCnt[t] = cnt;
    __syncthreads();

    // ---- Fixed-order tree reduction: deterministic across replays ----
    for (int off = THREADS >> 1; off > 0; off >>= 1) {
        if (t < off) {
            sSum[t] += sSum[t + off];
            sCnt[t] += sCnt[t + off];
        }
        __syncthreads();
    }

    if (t == 0) {
        const int seq_end = sCnt[0] - 1;           // sum(mask) - 1
        const int tag0    = sTags[0];
        const int tagL    = sTags[seq_end];
        out[b] = start_tr[tag0]
               + emissions[(size_t)b * NTAGS + tag0]   // emissions[0, b, tag0]
               + sSum[0]
               + end_tr[tagL];
    }
}

extern "C" void kernel_launch(void* const* d_in, const int* in_sizes, int n_in,
                              void* d_out, int out_size, void* d_ws, size_t ws_size,
                              hipStream_t stream) {
    (void)in_sizes; (void)n_in; (void)out_size; (void)d_ws; (void)ws_size;
    const float* emissions = (const float*)d_in[0];   // (512,512,128) f32
    const int*   tags      = (const int*)d_in[1];     // (512,512) int
    const int*   mask      = (const int*)d_in[2];     // (512,512) int
    const float* start_tr  = (const float*)d_in[3];   // (128,)
    const float* end_tr    = (const float*)d_in[4];   // (128,)
    const float* trans     = (const float*)d_in[5];   // (128,128)
    float*       out       = (float*)d_out;           // (512,)

    crf_score_kernel<<<BATCH, THREADS, 0, stream>>>(
        emissions, tags, mask, start_tr, end_tr, trans, out);
}